// GAFCrossAttnHead_49374944035447
// MI455X (gfx1250) — compile-verified
//
#include <hip/hip_runtime.h>
#include <math.h>

// Problem constants (match reference)
#define B_        128
#define D_TCF     512
#define T_        1024
#define GAF_DIM   256
#define DK        32
#define HIDDEN    64
#define NCLS      4
#define EPS_      1e-5f
#define SCALE_    0.17677669529663687f  // 32^-0.5

typedef float v2f __attribute__((ext_vector_type(2)));
typedef float v4f __attribute__((ext_vector_type(4)));
typedef float v8f __attribute__((ext_vector_type(8)));

// ---------------------------------------------------------------------------
// k1: per-batch tiny precompute.
//   KV[b,k]   = sum_g gaf[b,g]*Wkv[k,g]
//   u[b,d]    = sum_k KV[b,k]*Wout[d,k]       (attn value projected by Wout)
//   wqkv[b,d] = sum_k KV[b,k]*Wq[k,d]         (folds Q-proj into score matvec)
//   SU[b]=sum_d u, SUU[b]=sum_d u^2           (closed-form LN stats helpers)
// ---------------------------------------------------------------------------
__global__ void k1_prep(const float* __restrict__ gaf, const float* __restrict__ Wkv,
                        const float* __restrict__ Wout, const float* __restrict__ Wq,
                        float* __restrict__ u, float* __restrict__ wqkv,
                        float* __restrict__ SU, float* __restrict__ SUU) {
  __shared__ float kv_s[DK];
  __shared__ float red[256];
  const int b = blockIdx.x;
  const int tid = threadIdx.x;
  if (tid < DK) {
    float acc = 0.f;
    const float* g  = gaf + b * GAF_DIM;
    const float* wk = Wkv + tid * GAF_DIM;
    for (int i = 0; i < GAF_DIM; ++i) acc = fmaf(g[i], wk[i], acc);
    kv_s[tid] = acc;
  }
  __syncthreads();
  float su = 0.f, suu = 0.f;
  for (int d = tid; d < D_TCF; d += 256) {
    float ud = 0.f, wd = 0.f;
#pragma unroll
    for (int k = 0; k < DK; ++k) {
      const float kv = kv_s[k];
      ud = fmaf(kv, Wout[d * DK + k], ud);
      wd = fmaf(kv, Wq[k * D_TCF + d], wd);
    }
    u[b * D_TCF + d]    = ud;
    wqkv[b * D_TCF + d] = wd;
    su += ud;
    suu = fmaf(ud, ud, suu);
  }
  red[tid] = su; __syncthreads();
  for (int s = 128; s > 0; s >>= 1) { if (tid < s) red[tid] += red[tid + s]; __syncthreads(); }
  if (tid == 0) SU[b] = red[0];
  __syncthreads();
  red[tid] = suu; __syncthreads();
  for (int s = 128; s > 0; s >>= 1) { if (tid < s) red[tid] += red[tid + s]; __syncthreads(); }
  if (tid == 0) SUU[b] = red[0];
}

// ---------------------------------------------------------------------------
// k2: streaming pass 1 over tcf_feat (256 MB). For each (b,t):
//   dot = sum_d x*wqkv ; S1 = sum_d x ; S2 = sum_d x^2 ; S3 = sum_d x*u
// Each thread owns 2 consecutive t (b64 loads); block = 256 t;
// grid = 512 blocks (2048 waves) + global_prefetch_b8 stream 96 KB ahead.
// ---------------------------------------------------------------------------
__global__ void k2_passA(const float* __restrict__ tcf,
                         const float* __restrict__ u, const float* __restrict__ wqkv,
                         float* __restrict__ dot, float* __restrict__ S1,
                         float* __restrict__ S2, float* __restrict__ S3) {
  __shared__ float u_s[D_TCF];
  __shared__ float w_s[D_TCF];
  const int b  = blockIdx.x >> 2;                        // 4 tiles of 256 t per batch
  const int t0 = ((blockIdx.x & 3) << 8) + (threadIdx.x << 1);
  for (int i = threadIdx.x; i < D_TCF; i += 128) {
    u_s[i] = u[b * D_TCF + i];
    w_s[i] = wqkv[b * D_TCF + i];
  }
  __syncthreads();
  const float* base = tcf + (size_t)b * D_TCF * T_ + t0;
  float d0 = 0.f, d1 = 0.f, s10 = 0.f, s11 = 0.f;
  float s20 = 0.f, s21 = 0.f, s30 = 0.f, s31 = 0.f;
  for (int d = 0; d < D_TCF; ++d) {
    __builtin_prefetch(base + (size_t)(d + 24) * T_, 0, 0);  // global_prefetch_b8
    const v2f x = *(const v2f*)(base + (size_t)d * T_);
    const float wd = w_s[d];
    const float ud = u_s[d];
    d0 = fmaf(x.x, wd, d0);    d1 = fmaf(x.y, wd, d1);
    s10 += x.x;                s11 += x.y;
    s20 = fmaf(x.x, x.x, s20); s21 = fmaf(x.y, x.y, s21);
    s30 = fmaf(x.x, ud, s30);  s31 = fmaf(x.y, ud, s31);
  }
  const int idx = b * T_ + t0;
  *(v2f*)(dot + idx) = (v2f){d0, d1};
  *(v2f*)(S1 + idx)  = (v2f){s10, s11};
  *(v2f*)(S2 + idx)  = (v2f){s20, s21};
  *(v2f*)(S3 + idx)  = (v2f){s30, s31};
}

// ---------------------------------------------------------------------------
// k3: per-batch softmax over T + closed-form LayerNorm stats.
//   w = softmax(scale*dot); mu=(S1+w*SU)/512; var=(S2+2wS3+w^2 SUU)/512 - mu^2
//   r = rsqrt(var+eps);  A1[b]=sum_t r*w ; A2[b]=sum_t r*mu
// ---------------------------------------------------------------------------
__global__ void k3_softmax(const float* __restrict__ dot, const float* __restrict__ S1,
                           const float* __restrict__ S2, const float* __restrict__ S3,
                           const float* __restrict__ SU, const float* __restrict__ SUU,
                           float* __restrict__ r_out, float* __restrict__ A1,
                           float* __restrict__ A2) {
  __shared__ float red[T_];
  const int b = blockIdx.x;
  const int t = threadIdx.x;
  const int idx = b * T_ + t;
  const float score = dot[idx] * SCALE_;
  red[t] = score; __syncthreads();
  for (int s = T_ / 2; s > 0; s >>= 1) { if (t < s) red[t] = fmaxf(red[t], red[t + s]); __syncthreads(); }
  const float m = red[0]; __syncthreads();
  const float e = expf(score - m);
  red[t] = e; __syncthreads();
  for (int s = T_ / 2; s > 0; s >>= 1) { if (t < s) red[t] += red[t + s]; __syncthreads(); }
  const float denom = red[0]; __syncthreads();
  const float w = e / denom;
  const float su = SU[b], suu = SUU[b];
  const float mu  = (S1[idx] + w * su) * (1.f / D_TCF);
  const float msq = (S2[idx] + 2.f * w * S3[idx] + w * w * suu) * (1.f / D_TCF);
  const float var = msq - mu * mu;
  const float r = rsqrtf(var + EPS_);
  r_out[idx] = r;
  red[t] = r * w; __syncthreads();
  for (int s = T_ / 2; s > 0; s >>= 1) { if (t < s) red[t] += red[t + s]; __syncthreads(); }
  if (t == 0) A1[b] = red[0];
  __syncthreads();
  red[t] = r * mu; __syncthreads();
  for (int s = T_ / 2; s > 0; s >>= 1) { if (t < s) red[t] += red[t + s]; __syncthreads(); }
  if (t == 0) A2[b] = red[0];
}

// ---------------------------------------------------------------------------
// k4: streaming pass 2 over tcf_feat. One wave per (b,d) row, b128 loads:
//   pooled[b,d] = g[d]/T * ( sum_t r[b,t]*x[b,d,t] + u[b,d]*A1[b] - A2[b] ) + beta[d]
// ---------------------------------------------------------------------------
__global__ void k4_pool(const float* __restrict__ tcf, const float* __restrict__ r,
                        const float* __restrict__ u, const float* __restrict__ A1,
                        const float* __restrict__ A2, const float* __restrict__ ln1_g,
                        const float* __restrict__ ln1_b, float* __restrict__ pooled) {
  __shared__ float r_s[T_];
  const int b = blockIdx.x >> 6;                       // 64 blocks of 8 rows per batch
  const int d = ((blockIdx.x & 63) << 3) + (threadIdx.x >> 5);
  const int lane = threadIdx.x & 31;
  for (int i = threadIdx.x << 2; i < T_; i += 1024)    // v4 stage of r row
    *(v4f*)(&r_s[i]) = *(const v4f*)(r + b * T_ + i);
  __syncthreads();
  const float* row = tcf + (size_t)b * D_TCF * T_ + (size_t)d * T_;
  float s = 0.f;
  for (int c = 0; c < T_; c += 128) {                  // 8 iters of b128 per lane
    const v4f x  = *(const v4f*)(row + c + (lane << 2));
    const v4f rv = *(const v4f*)(&r_s[c + (lane << 2)]);
    s = fmaf(x.x, rv.x, s);
    s = fmaf(x.y, rv.y, s);
    s = fmaf(x.z, rv.z, s);
    s = fmaf(x.w, rv.w, s);
  }
  for (int off = 16; off > 0; off >>= 1) s += __shfl_down(s, off, 32);
  if (lane == 0) {
    const float val = (s + u[b * D_TCF + d] * A1[b] - A2[b]) * (1.f / T_);
    pooled[b * D_TCF + d] = fmaf(ln1_g[d], val, ln1_b[d]);
  }
}

// ---------------------------------------------------------------------------
// k5: Z = pooled[128,512] @ W1^T[512,64]  via V_WMMA_F32_16X16X4_F32.
// One wave per 16x16 output tile; K=512 -> 128 WMMA steps.
// fp32 layouts (ISA 7.12.2):
//   A: lane l, elem v  -> A[l%16, 2*(l/16)+v]
//   B: lane l, elem v  -> B[2*(l/16)+v, l%16]
//   C/D: lane l, vgpr v -> C[v + 8*(l/16), l%16]
// ---------------------------------------------------------------------------
__global__ void k5_wmma(const float* __restrict__ pooled, const float* __restrict__ W1,
                        float* __restrict__ Z) {
  const int lane = threadIdx.x;            // 32 threads = 1 wave per block
  const int mt = blockIdx.x & 7;           // 8 tiles of M=128
  const int nt = blockIdx.x >> 3;          // 4 tiles of N=64
  const int half = lane >> 4;
  const int l16  = lane & 15;
  const int rowA = mt * 16 + l16;          // row of pooled
  const int rowB = nt * 16 + l16;          // row of W1 (N index)
  const int koff = half * 2;
  const float* pa = pooled + rowA * D_TCF + koff;
  const float* pb = W1 + rowB * D_TCF + koff;
  v8f acc = {};
  for (int k = 0; k < D_TCF; k += 4) {
    const v2f a  = *(const v2f*)(pa + k);
    const v2f bb = *(const v2f*)(pb + k);
    // (neg_a, A, neg_b, B, c_mod, C, reuse_a, reuse_b)
    acc = __builtin_amdgcn_wmma_f32_16x16x4_f32(false, a, false, bb, (short)0, acc,
                                                false, false);
  }
#pragma unroll
  for (int v = 0; v < 8; ++v) {
    const int row = mt * 16 + v + half * 8;
    const int col = nt * 16 + l16;
    Z[row * HIDDEN + col] = acc[v];
  }
}

// ---------------------------------------------------------------------------
// k6: out = (elu(LN2(Z + b1))) @ W2^T + b2
// ---------------------------------------------------------------------------
__global__ void k6_head(const float* __restrict__ Z, const float* __restrict__ b1,
                        const float* __restrict__ ln2_g, const float* __restrict__ ln2_b,
                        const float* __restrict__ W2, const float* __restrict__ b2,
                        float* __restrict__ out) {
  __shared__ float red[HIDDEN];
  __shared__ float h_s[HIDDEN];
  const int b = blockIdx.x;
  const int j = threadIdx.x;
  const float z = Z[b * HIDDEN + j] + b1[j];
  red[j] = z; __syncthreads();
  for (int s = HIDDEN / 2; s > 0; s >>= 1) { if (j < s) red[j] += red[j + s]; __syncthreads(); }
  const float mu = red[0] * (1.f / HIDDEN); __syncthreads();
  const float dv = z - mu;
  red[j] = dv * dv; __syncthreads();
  for (int s = HIDDEN / 2; s > 0; s >>= 1) { if (j < s) red[j] += red[j + s]; __syncthreads(); }
  const float var = red[0] * (1.f / HIDDEN); __syncthreads();
  const float rr = rsqrtf(var + EPS_);
  const float hn = fmaf(dv * rr, ln2_g[j], ln2_b[j]);
  h_s[j] = hn > 0.f ? hn : (expf(hn) - 1.f);   // ELU, alpha=1
  __syncthreads();
  if (j < NCLS) {
    float o = b2[j];
#pragma unroll
    for (int k = 0; k < HIDDEN; ++k) o = fmaf(h_s[k], W2[j * HIDDEN + k], o);
    out[b * NCLS + j] = o;
  }
}

extern "C" void kernel_launch(void* const* d_in, const int* in_sizes, int n_in,
                              void* d_out, int out_size, void* d_ws, size_t ws_size,
                              hipStream_t stream) {
  const float* tcf  = (const float*)d_in[0];
  const float* gaf  = (const float*)d_in[1];
  const float* Wq   = (const float*)d_in[2];
  const float* Wkv  = (const float*)d_in[3];
  const float* Wout = (const float*)d_in[4];
  const float* ln1g = (const float*)d_in[5];
  const float* ln1b = (const float*)d_in[6];
  const float* W1   = (const float*)d_in[7];
  const float* b1   = (const float*)d_in[8];
  const float* ln2g = (const float*)d_in[9];
  const float* ln2b = (const float*)d_in[10];
  const float* W2   = (const float*)d_in[11];
  const float* b2   = (const float*)d_in[12];
  float* out = (float*)d_out;

  // Workspace carve-up (floats): ~3.44 MB total
  float* ws     = (float*)d_ws;
  float* u      = ws;                    // B*D = 65536
  float* wqkv   = u + B_ * D_TCF;        // 65536
  float* SU     = wqkv + B_ * D_TCF;     // 128
  float* SUU    = SU + B_;               // 128
  float* dot    = SUU + B_;              // B*T = 131072
  float* S1     = dot + B_ * T_;
  float* S2     = S1 + B_ * T_;
  float* S3     = S2 + B_ * T_;
  float* rbt    = S3 + B_ * T_;
  float* A1     = rbt + B_ * T_;         // 128
  float* A2     = A1 + B_;               // 128
  float* pooled = A2 + B_;               // 65536
  float* Z      = pooled + B_ * D_TCF;   // 8192

  k1_prep<<<B_, 256, 0, stream>>>(gaf, Wkv, Wout, Wq, u, wqkv, SU, SUU);
  k2_passA<<<B_ * (T_ / 256), 128, 0, stream>>>(tcf, u, wqkv, dot, S1, S2, S3);
  k3_softmax<<<B_, T_, 0, stream>>>(dot, S1, S2, S3, SU, SUU, rbt, A1, A2);
  k4_pool<<<B_ * (D_TCF / 8), 256, 0, stream>>>(tcf, rbt, u, A1, A2, ln1g, ln1b, pooled);
  k5_wmma<<<32, 32, 0, stream>>>(pooled, W1, Z);
  k6_head<<<B_, HIDDEN, 0, stream>>>(Z, b1, ln2g, ln2b, W2, b2, out);
}